// GTLayer_44349832298688
// MI455X (gfx1250) — compile-verified
//
#include <hip/hip_runtime.h>
#include <math.h>

// ---------------------------------------------------------------------------
// Graph-Transformer layer for MI455X (gfx1250, wave32, WMMA).
// Node-level QKV GEMM via V_WMMA_F32_16X16X4_F32, async global->LDS staging
// (ASYNCcnt path) when available, edge phase as L2-resident gathers + fp32
// global atomics, wave32 LayerNorm.
// ---------------------------------------------------------------------------

typedef __attribute__((ext_vector_type(2))) float v2f;
typedef __attribute__((ext_vector_type(8))) float v8f;

#define D_MODEL 128
#define N_HEAD 8
#define HEAD_DIM 16

#if __has_builtin(__builtin_amdgcn_global_load_async_to_lds_b128) && \
    __has_builtin(__builtin_amdgcn_s_wait_asynccnt)
#define USE_ASYNC_LDS 1
// Parameter types per hipcc diagnostic: pointer to GCC int4 vector; global
// source lives in AS1 (printed as "__device__"), LDS dest in AS3.
typedef int v4i_gcc __attribute__((__vector_size__(16)));
typedef __attribute__((address_space(1))) v4i_gcc gas_v4i;
typedef __attribute__((address_space(3))) v4i_gcc las_v4i;
#endif

// ---------------------------------------------------------------------------
// Zero-fill kernel (res accumulator + attNorm must start at 0 every call).
// ---------------------------------------------------------------------------
__global__ void zero_f32_kernel(float* __restrict__ p, long long n) {
    long long i = (long long)blockIdx.x * blockDim.x + threadIdx.x;
    if (i < n) p[i] = 0.0f;
}

// ---------------------------------------------------------------------------
// Kernel 1: Q = E @ Wq, K = E @ Wk, V = E @ Wv  (node-level, fp32 WMMA).
// Block = 256 threads = 8 waves. Block covers 16 node rows; wave w owns the
// 16-column output tile [w*16, w*16+16). A (16x128 embeds slab) is staged in
// LDS once (via GLOBAL_LOAD_ASYNC_TO_LDS_B128 when available) and shared by
// all three GEMMs. K-loop: 32 steps of K=4.
//
// WMMA f32 16x16x4 operand layout (ISA 7.12.2):
//   A (16x4, MxK): lane L<16  -> {A[L][k0+0], A[L][k0+1]}
//                  lane L>=16 -> {A[L-16][k0+2], A[L-16][k0+3]}
//   B (4x16, KxN): mirrored: lane L<16  -> {B[k0+0][L], B[k0+1][L]}
//                  lane L>=16 -> {B[k0+2][L-16], B[k0+3][L-16]}
//   C/D (16x16):   VGPR r: lanes 0-15 = row r, lanes 16-31 = row r+8.
// ---------------------------------------------------------------------------
__global__ void __launch_bounds__(256)
qkv_gemm_kernel(const float* __restrict__ embeds,
                const float* __restrict__ Wq,
                const float* __restrict__ Wk,
                const float* __restrict__ Wv,
                float* __restrict__ Q,
                float* __restrict__ K,
                float* __restrict__ V)
{
    __shared__ float sA[16 * D_MODEL];   // 8 KB of the 320 KB WGP LDS

    const int       tid  = threadIdx.x;          // 0..255
    const long long row0 = (long long)blockIdx.x * 16;

    // Stage the 16x128 fp32 embeds slab into LDS (2048 floats, 512 x b128).
#ifdef USE_ASYNC_LDS
    {
        const float4* src = (const float4*)(embeds + row0 * D_MODEL);
        float4*       dst = (float4*)sA;
        __builtin_amdgcn_global_load_async_to_lds_b128(
            (gas_v4i*)(src + tid), (las_v4i*)(dst + tid), 0, 0);
        __builtin_amdgcn_global_load_async_to_lds_b128(
            (gas_v4i*)(src + tid + 256), (las_v4i*)(dst + tid + 256), 0, 0);
        __builtin_amdgcn_s_wait_asynccnt(0);   // own wave's copies landed
    }
#else
    {
        const float4* src = (const float4*)(embeds + row0 * D_MODEL);
        float4*       dst = (float4*)sA;
        dst[tid]       = src[tid];
        dst[tid + 256] = src[tid + 256];
    }
#endif
    __syncthreads();                            // cross-wave visibility

    const int wave  = tid >> 5;          // 0..7 -> output column tile
    const int lane  = tid & 31;
    const int m     = lane & 15;         // row within tile / col within tile
    const int kh    = (lane >> 4) << 1;  // 0 for lanes 0-15, 2 for lanes 16-31
    const int nbase = wave * 16;
    const int col   = nbase + m;

    v8f cq = {}; v8f ck = {}; v8f cv = {};

    for (int kk = 0; kk < D_MODEL; kk += 4) {
        const int krow = kk + kh;

        // A fragment from LDS (ds_load path)
        v2f a;
        a.x = sA[m * D_MODEL + krow];
        a.y = sA[m * D_MODEL + krow + 1];

        // B fragments straight from global; the three 64 KB weight matrices
        // are L2/WGP$-resident after the first block touches them.
        v2f bq, bk, bv;
        bq.x = Wq[krow * D_MODEL + col];
        bq.y = Wq[(krow + 1) * D_MODEL + col];
        bk.x = Wk[krow * D_MODEL + col];
        bk.y = Wk[(krow + 1) * D_MODEL + col];
        bv.x = Wv[krow * D_MODEL + col];
        bv.y = Wv[(krow + 1) * D_MODEL + col];

        // D = A*B + C, fp32 end to end (matches reference precision).
        cq = __builtin_amdgcn_wmma_f32_16x16x4_f32(false, a, false, bq, (short)0, cq, false, false);
        ck = __builtin_amdgcn_wmma_f32_16x16x4_f32(false, a, false, bk, (short)0, ck, false, false);
        cv = __builtin_amdgcn_wmma_f32_16x16x4_f32(false, a, false, bv, (short)0, cv, false, false);
    }

    // Write back per the C/D layout: VGPR r -> rows r (lanes 0-15) / r+8.
    const long long srow = row0 + (long long)((lane >> 4) * 8);
    #pragma unroll
    for (int r = 0; r < 8; ++r) {
        const long long off = (srow + r) * D_MODEL + col;
        Q[off] = cq[r];
        K[off] = ck[r];
        V[off] = cv[r];
    }
}

// ---------------------------------------------------------------------------
// Kernel 2: per (edge, head) attention logits -> exp -> segment-sum (atomic).
// Q/K are L2-resident (25.6 MB each on a 192 MB L2).
// ---------------------------------------------------------------------------
__global__ void __launch_bounds__(256)
edge_scores_kernel(const float* __restrict__ Q,
                   const float* __restrict__ K,
                   const long long* __restrict__ rows,
                   const long long* __restrict__ cols,
                   float* __restrict__ expAtt,
                   float* __restrict__ attNorm,
                   long long n_edges)
{
    long long idx = (long long)blockIdx.x * blockDim.x + threadIdx.x;
    if (idx >= n_edges * N_HEAD) return;

    const long long e = idx >> 3;
    const int       h = (int)(idx & 7);
    const long long r = rows[e];
    const long long c = cols[e];

    const float4* q = (const float4*)(Q + r * D_MODEL + h * HEAD_DIM);
    const float4* k = (const float4*)(K + c * D_MODEL + h * HEAD_DIM);

    float s = 0.0f;
    #pragma unroll
    for (int i = 0; i < 4; ++i) {
        float4 a = q[i], b = k[i];
        s = fmaf(a.x, b.x, s); s = fmaf(a.y, b.y, s);
        s = fmaf(a.z, b.z, s); s = fmaf(a.w, b.w, s);
    }
    s = fminf(10.0f, fmaxf(-10.0f, s));
    float ex = expf(s);

    expAtt[idx] = ex;
    atomicAdd(&attNorm[r * N_HEAD + h], ex);   // global_atomic_add_f32
}

// ---------------------------------------------------------------------------
// Kernel 3: normalize + weighted-V scatter-add into res (fp32 atomics, L2).
// ---------------------------------------------------------------------------
__global__ void __launch_bounds__(256)
edge_aggregate_kernel(const float* __restrict__ V,
                      const float* __restrict__ expAtt,
                      const float* __restrict__ attNorm,
                      const long long* __restrict__ rows,
                      const long long* __restrict__ cols,
                      float* __restrict__ res,
                      long long n_edges)
{
    long long idx = (long long)blockIdx.x * blockDim.x + threadIdx.x;
    if (idx >= n_edges * N_HEAD) return;

    const long long e = idx >> 3;
    const int       h = (int)(idx & 7);
    const long long r = rows[e];
    const long long c = cols[e];

    const float w = expAtt[idx] / (attNorm[r * N_HEAD + h] + 1e-8f);

    const float4* v   = (const float4*)(V + c * D_MODEL + h * HEAD_DIM);
    float*        out = res + r * D_MODEL + h * HEAD_DIM;

    #pragma unroll
    for (int i = 0; i < 4; ++i) {
        float4 vv = v[i];
        atomicAdd(out + 4 * i + 0, w * vv.x);
        atomicAdd(out + 4 * i + 1, w * vv.y);
        atomicAdd(out + 4 * i + 2, w * vv.z);
        atomicAdd(out + 4 * i + 3, w * vv.w);
    }
}

// ---------------------------------------------------------------------------
// Kernel 4: residual + LayerNorm. One wave32 per node: 128 elems = 1 float4
// per lane; mean/var via __shfl_xor butterfly reduction (wave32-safe).
// ---------------------------------------------------------------------------
__global__ void __launch_bounds__(256)
layernorm_kernel(const float* __restrict__ res,
                 const float* __restrict__ embeds,
                 const float* __restrict__ ln_scale,
                 const float* __restrict__ ln_bias,
                 float* __restrict__ out,
                 long long n_nodes)
{
    const int       lane = threadIdx.x & 31;
    const long long node = (long long)blockIdx.x * (blockDim.x >> 5)
                         + (threadIdx.x >> 5);
    if (node >= n_nodes) return;

    const float4* rp = (const float4*)(res    + node * D_MODEL);
    const float4* ep = (const float4*)(embeds + node * D_MODEL);
    float4 x = rp[lane];
    float4 e = ep[lane];
    x.x += e.x; x.y += e.y; x.z += e.z; x.w += e.w;

    float sum = x.x + x.y + x.z + x.w;
    float sq  = x.x * x.x + x.y * x.y + x.z * x.z + x.w * x.w;

    #pragma unroll
    for (int off = 16; off >= 1; off >>= 1) {
        sum += __shfl_xor(sum, off, 32);
        sq  += __shfl_xor(sq,  off, 32);
    }

    const float mu   = sum * (1.0f / D_MODEL);
    const float var  = sq * (1.0f / D_MODEL) - mu * mu;
    const float rstd = rsqrtf(var + 1e-6f);

    const float4* sp = (const float4*)ln_scale;
    const float4* bp = (const float4*)ln_bias;
    float4 s4 = sp[lane];
    float4 b4 = bp[lane];

    float4 o;
    o.x = (x.x - mu) * rstd * s4.x + b4.x;
    o.y = (x.y - mu) * rstd * s4.y + b4.y;
    o.z = (x.z - mu) * rstd * s4.z + b4.z;
    o.w = (x.w - mu) * rstd * s4.w + b4.w;

    ((float4*)(out + node * D_MODEL))[lane] = o;
}

// ---------------------------------------------------------------------------
// Host-side launcher. All launches on `stream`; no alloc/sync (graph-safe).
// Workspace layout (floats): Q | K | V | res | attNorm | expAtt  (~130 MB).
// ---------------------------------------------------------------------------
extern "C" void kernel_launch(void* const* d_in, const int* in_sizes, int n_in,
                              void* d_out, int out_size, void* d_ws, size_t ws_size,
                              hipStream_t stream) {
    const float*     embeds   = (const float*)d_in[0];
    const long long* eidx     = (const long long*)d_in[1];   // int64 [2, E]
    const float*     Wq       = (const float*)d_in[2];
    const float*     Wk       = (const float*)d_in[3];
    const float*     Wv       = (const float*)d_in[4];
    const float*     ln_scale = (const float*)d_in[5];
    const float*     ln_bias  = (const float*)d_in[6];
    float*           out      = (float*)d_out;

    const long long n_nodes = in_sizes[0] / D_MODEL;   // 50000
    const long long n_edges = in_sizes[1] / 2;         // 800000

    const long long* rows = eidx;
    const long long* cols = eidx + n_edges;

    float* ws = (float*)d_ws;
    const long long nd = n_nodes * D_MODEL;
    float* Q       = ws;
    float* K       = ws + nd;
    float* V       = ws + 2 * nd;
    float* res     = ws + 3 * nd;
    float* attNorm = ws + 4 * nd;                      // n_nodes * N_HEAD
    float* expAtt  = attNorm + n_nodes * N_HEAD;       // n_edges * N_HEAD

    // 1) zero res + attNorm (contiguous in ws)
    {
        const long long nz = nd + n_nodes * N_HEAD;
        const int grid = (int)((nz + 255) / 256);
        zero_f32_kernel<<<grid, 256, 0, stream>>>(res, nz);
    }

    // 2) node-level QKV GEMM (N % 16 == 0 for N=50000 -> 3125 blocks)
    {
        const int grid = (int)(n_nodes / 16);
        qkv_gemm_kernel<<<grid, 256, 0, stream>>>(embeds, Wq, Wk, Wv, Q, K, V);
    }

    // 3) edge attention scores + segment-sum denominator
    {
        const long long total = n_edges * N_HEAD;
        const int grid = (int)((total + 255) / 256);
        edge_scores_kernel<<<grid, 256, 0, stream>>>(Q, K, rows, cols,
                                                     expAtt, attNorm, n_edges);
    }

    // 4) normalized weighted-V scatter into res
    {
        const long long total = n_edges * N_HEAD;
        const int grid = (int)((total + 255) / 256);
        edge_aggregate_kernel<<<grid, 256, 0, stream>>>(V, expAtt, attNorm,
                                                        rows, cols, res, n_edges);
    }

    // 5) residual + LayerNorm (8 nodes per 256-thread block, wave32 each)
    {
        const int grid = (int)((n_nodes + 7) / 8);
        layernorm_kernel<<<grid, 256, 0, stream>>>(res, embeds, ln_scale,
                                                   ln_bias, out, n_nodes);
    }
}